// MixedMSEPoweImbalance_773094113348
// MI455X (gfx1250) — compile-verified
//
#include <hip/hip_runtime.h>
#include <hip/hip_bf16.h>

typedef float v2f __attribute__((ext_vector_type(2)));
typedef float v8f __attribute__((ext_vector_type(8)));

#define DEG2RAD 0.017453292519943295f

// ---------------- zero workspace ----------------
__global__ void zero_ws_kernel(float* __restrict__ ws, int n) {
    int i = blockIdx.x * blockDim.x + threadIdx.x;
    if (i < n) ws[i] = 0.0f;
}

// ---------------- edge scatter kernel ----------------
// One thread per undirected edge; computes both directed messages and
// scatter-adds P,Q at src and dst via L2 float atomics.
__global__ void edge_kernel(const float*      __restrict__ x,
                            const long long*  __restrict__ eidx,   // (2,E) int64
                            const float*      __restrict__ eattr,  // (E,2)
                            const float*      __restrict__ xymean,
                            const float*      __restrict__ xystd,
                            const float*      __restrict__ edgemean,
                            const float*      __restrict__ edgestd,
                            float*            __restrict__ agg,    // (N,2)
                            int E) {
    int e = blockIdx.x * blockDim.x + threadIdx.x;
    if (e >= E) return;

    // Stream-ahead prefetch of the three coalesced edge streams
    // (emits global_prefetch_b8; speculative, safe at the tail).
    __builtin_prefetch(&eidx[e + 8192]);
    __builtin_prefetch(&eidx[(size_t)E + e + 8192]);
    __builtin_prefetch(&eattr[2 * (e + 8192)]);

    int s = (int)eidx[e];
    int d = (int)eidx[(size_t)E + e];

    float r  = fmaf(eattr[2 * e],     edgestd[0], edgemean[0]);
    float xr = fmaf(eattr[2 * e + 1], edgestd[1], edgemean[1]);
    float inv = 1.0f / fmaf(r, r, xr * xr);
    float g =  r  * inv;
    float b = -xr * inv;

    float xm0 = xymean[0], xm1 = xymean[1];
    float xs0 = xystd[0],  xs1 = xystd[1];

    // x table is 2.4 MB -> L2-resident gathers
    float vms = fmaf(x[6 * (size_t)s],     xs0, xm0);
    float vas = fmaf(x[6 * (size_t)s + 1], xs1, xm1) * DEG2RAD;
    float vmd = fmaf(x[6 * (size_t)d],     xs0, xm0);
    float vad = fmaf(x[6 * (size_t)d + 1], xs1, xm1) * DEG2RAD;

    float ss, cs, sd_, cd;
    __sincosf(vas, &ss, &cs);
    __sincosf(vad, &sd_, &cd);
    float es = vms * cs, fs = vms * ss;
    float ed = vmd * cd, fd = vmd * sd_;

    // shared subexpressions for both directions
    float c     = fmaf(es, ed, fs * fd);   // e_i e_j + f_i f_j
    float cross = fmaf(fs, ed, -es * fd);  // f_i e_j - e_i f_j (i=src)
    float ns = vms * vms;                  // e_s^2 + f_s^2
    float nd = vmd * vmd;

    float t1 = c - ns;                     // i=src, j=dst
    float u1 = c - nd;                     // i=dst, j=src  (cross flips sign)

    float P1 = fmaf(g, t1,  b * cross);
    float Q1 = fmaf(g, cross, -b * t1);
    float P2 = fmaf(g, u1, -b * cross);
    float Q2 = fmaf(-g, cross, -b * u1);

    atomicAdd(&agg[2 * (size_t)s],     P1);
    atomicAdd(&agg[2 * (size_t)s + 1], Q1);
    atomicAdd(&agg[2 * (size_t)d],     P2);
    atomicAdd(&agg[2 * (size_t)d + 1], Q2);
}

// ---------------- WMMA wave reduction ----------------
// Sum 32 per-lane f32 values with one V_WMMA_F32_16X16X4_F32.
// A: VGPR0 = val (K=0 for lanes 0-15, K=2 for lanes 16-31), VGPR1 = 0
// B: all-ones 4x16 (layout-independent) => D[m][n] = val(m) + val(m+16).
// Sum the 8 D VGPRs -> lanes 0-15 hold sum(rows 0..7), lanes 16-31 sum(rows 8..15);
// one shfl_xor(16) gives the full 32-lane total in every lane. Exact f32 math.
__device__ __forceinline__ float wave_reduce_wmma(float val) {
    v2f a;  a[0] = val;  a[1] = 0.0f;
    v2f bm; bm[0] = 1.0f; bm[1] = 1.0f;
    v8f cz = {};
    v8f dm = __builtin_amdgcn_wmma_f32_16x16x4_f32(
        /*neg_a=*/false, a, /*neg_b=*/false, bm,
        /*c_mod=*/(short)0, cz, /*reuse_a=*/false, /*reuse_b=*/false);
    float t = ((dm[0] + dm[1]) + (dm[2] + dm[3])) +
              ((dm[4] + dm[5]) + (dm[6] + dm[7]));
    t += __shfl_xor(t, 16, 32);
    return t;
}

// ---------------- node kernel: dPQ + MSE partial sums ----------------
__global__ void node_kernel(const float* __restrict__ x,
                            const float* __restrict__ y,
                            const float* __restrict__ xymean,
                            const float* __restrict__ xystd,
                            const float* __restrict__ agg,
                            float*       __restrict__ acc,   // acc[0]=sum dPQ, acc[1]=sum MSE
                            int N) {
    int n = blockIdx.x * blockDim.x + threadIdx.x;
    float dpq = 0.0f, mse = 0.0f;
    if (n < N) {
        float xd2 = fmaf(x[6 * (size_t)n + 2], xystd[2], xymean[2]);
        float xd3 = fmaf(x[6 * (size_t)n + 3], xystd[3], xymean[3]);
        float dP = xd2 - agg[2 * (size_t)n];
        float dQ = xd3 - agg[2 * (size_t)n + 1];
        dpq = fmaf(dP, dP, dQ * dQ);
#pragma unroll
        for (int c = 0; c < 6; ++c) {
            float df = x[6 * (size_t)n + c] - y[6 * (size_t)n + c];
            mse = fmaf(df, df, mse);
        }
    }
    // all 32 lanes converged here (EXEC all-ones) -> WMMA legal
    float wdpq = wave_reduce_wmma(dpq);
    float wmse = wave_reduce_wmma(mse);
    if ((threadIdx.x & 31) == 0) {
        atomicAdd(&acc[0], wdpq);
        atomicAdd(&acc[1], wmse);
    }
}

// ---------------- finalize ----------------
__global__ void finalize_kernel(const float* __restrict__ acc,
                                float* __restrict__ out, int N) {
    float power_imb = acc[0] / (float)N;
    float mse       = acc[1] / (float)(6 * N);
    out[0] = 0.5f * mse + 0.5f * 0.02f * power_imb;
}

extern "C" void kernel_launch(void* const* d_in, const int* in_sizes, int n_in,
                              void* d_out, int out_size, void* d_ws, size_t ws_size,
                              hipStream_t stream) {
    const float*     x        = (const float*)d_in[0];
    const long long* eidx     = (const long long*)d_in[1];  // int64
    const float*     eattr    = (const float*)d_in[2];
    const float*     y        = (const float*)d_in[3];
    const float*     xymean   = (const float*)d_in[4];
    const float*     xystd    = (const float*)d_in[5];
    const float*     edgemean = (const float*)d_in[6];
    const float*     edgestd  = (const float*)d_in[7];

    const int N = in_sizes[0] / 6;      // 100000
    const int E = in_sizes[1] / 2;      // 1600000

    float* agg = (float*)d_ws;               // 2*N floats (800 KB, L2-resident)
    float* acc = agg + (size_t)2 * N;        // 2 floats

    const int zn = 2 * N + 2;
    zero_ws_kernel<<<(zn + 255) / 256, 256, 0, stream>>>(agg, zn);

    edge_kernel<<<(E + 255) / 256, 256, 0, stream>>>(
        x, eidx, eattr, xymean, xystd, edgemean, edgestd, agg, E);

    node_kernel<<<(N + 255) / 256, 256, 0, stream>>>(
        x, y, xymean, xystd, agg, acc, N);

    finalize_kernel<<<1, 1, 0, stream>>>(acc, (float*)d_out, N);
}